// GCN_64390149702046
// MI455X (gfx1250) — compile-verified
//
#include <hip/hip_runtime.h>

#define N_NODES 100000
#define N_EDGES 1600000
#define F_INPUT 128
#define HID 32
#define N_GRAPHS 512

typedef __attribute__((ext_vector_type(2))) float v2f;
typedef __attribute__((ext_vector_type(8))) float v8f;

// ---------------- utility: vectorized zero ----------------
__global__ void zero_kernel(float4* __restrict__ p, long n4) {
  long i = (long)blockIdx.x * blockDim.x + threadIdx.x;
  long stride = (long)gridDim.x * blockDim.x;
  const float4 z = {0.f, 0.f, 0.f, 0.f};
  for (; i < n4; i += stride) p[i] = z;
}

// deg[col[e]] += 1
__global__ void degree_kernel(const int* __restrict__ col, float* __restrict__ deg, int E) {
  int e = blockIdx.x * blockDim.x + threadIdx.x;
  if (e < E) atomicAdd(&deg[col[e]], 1.0f);
}

// dinv = 1/sqrt(deg + 1)  (in place)
__global__ void dinv_kernel(float* __restrict__ deg, int n) {
  int i = blockIdx.x * blockDim.x + threadIdx.x;
  if (i < n) deg[i] = rsqrtf(deg[i] + 1.0f);
}

// ---------------- WMMA GEMM: Hout[M,32] = X[M,K] @ W[K,32] ----------------
// One wave computes a 16x32 tile using V_WMMA_F32_16X16X4_F32.
// A (16x4 f32): lanes 0-15 rows M=0..15 holding K={kb,kb+1}; lanes 16-31 K={kb+2,kb+3}.
// B (4x16 f32): lanes 0-15 cols N=0..15 holding K={kb,kb+1}; lanes 16-31 K={kb+2,kb+3}.
// C/D (16x16 f32): VGPR i -> M=i (lanes 0-15), M=8+i (lanes 16-31); N = lane&15.
template <int K>
__global__ void gemm_wmma_kernel(const float* __restrict__ X, const float* __restrict__ W,
                                 float* __restrict__ Hout, int M) {
  const int wave = (int)((blockIdx.x * blockDim.x + threadIdx.x) >> 5);
  const int lane = threadIdx.x & 31;
  const int row0 = wave * 16;
  if (row0 >= M) return;  // whole-wave exit: WMMA waves keep EXEC all-ones

  const int mrow = row0 + (lane & 15);
  const int koff = (lane >> 4) * 2;  // 0 or 2
  const int ncol = (lane & 15);

  v8f c0 = {};
  v8f c1 = {};

#pragma unroll 4
  for (int kb = 0; kb < K; kb += 4) {
    const int k = kb + koff;
    v2f a;
    a.x = X[(long)mrow * K + k];
    a.y = X[(long)mrow * K + k + 1];
    v2f b0, b1;
    b0.x = W[(long)k * HID + ncol];
    b0.y = W[(long)(k + 1) * HID + ncol];
    b1.x = W[(long)k * HID + ncol + 16];
    b1.y = W[(long)(k + 1) * HID + ncol + 16];
    c0 = __builtin_amdgcn_wmma_f32_16x16x4_f32(false, a, false, b0, (short)0, c0, false, false);
    c1 = __builtin_amdgcn_wmma_f32_16x16x4_f32(false, a, false, b1, (short)0, c1, false, false);
  }

  const int rbase = row0 + ((lane >> 4) * 8);
#pragma unroll
  for (int i = 0; i < 8; ++i) {
    Hout[(long)(rbase + i) * HID + ncol]      = c0[i];
    Hout[(long)(rbase + i) * HID + ncol + 16] = c1[i];
  }
}

// ---------------- edge scatter: agg[col] += dinv[row]*dinv[col] * h[row] ----------------
// One wave per edge; lane == feature (wave32 == HID).
// Edge id / indices / norms are wave-uniform -> force scalar (SMEM) path.
__global__ void edge_scatter_kernel(const int* __restrict__ row, const int* __restrict__ col,
                                    const float* __restrict__ dinv,
                                    const float* __restrict__ Hlin,
                                    float* __restrict__ agg, int E) {
  const int ew = (int)((blockIdx.x * blockDim.x + threadIdx.x) >> 5);
  const int f = threadIdx.x & 31;
  const int e = __builtin_amdgcn_readfirstlane(ew);
  if (e >= E) return;
  // prefetch the index stream a few waves ahead (speculative: OOB is dropped)
  __builtin_prefetch(&row[e + 128], 0, 1);
  __builtin_prefetch(&col[e + 128], 0, 1);
  const int r = __builtin_amdgcn_readfirstlane(row[e]);
  const int c = __builtin_amdgcn_readfirstlane(col[e]);
  const float w = dinv[r] * dinv[c];
  atomicAdd(&agg[(long)c * HID + f], w * Hlin[(long)r * HID + f]);
}

// ---------------- activation: out = relu(agg + dinv^2 * Hlin + b)  (in place on agg) ----
// float4 per thread (HID==32 -> 8 quads per node).
__global__ void act_kernel(const float4* __restrict__ Hlin, const float* __restrict__ dinv,
                           const float4* __restrict__ bias, float4* __restrict__ agg, int n) {
  const long q = (long)blockIdx.x * blockDim.x + threadIdx.x;
  if (q >= (long)n * (HID / 4)) return;
  const int i = (int)(q >> 3);  // node index
  const float d = dinv[i];
  const float dd = d * d;
  const float4 h = Hlin[q];
  const float4 b = bias[(int)(q & 7)];
  float4 a = agg[q];
  a.x = fmaxf(a.x + dd * h.x + b.x, 0.0f);
  a.y = fmaxf(a.y + dd * h.y + b.y, 0.0f);
  a.z = fmaxf(a.z + dd * h.z + b.z, 0.0f);
  a.w = fmaxf(a.w + dd * h.w + b.w, 0.0f);
  agg[q] = a;
}

// ---------------- pooling: per-graph max + sum + counts ----------------
// One wave per node; lane == feature. act >= 0 so float max == uint max on bits.
__global__ void pool_kernel(const float* __restrict__ act, const int* __restrict__ batch,
                            float* __restrict__ gmax, float* __restrict__ gsum,
                            float* __restrict__ cnt, int n) {
  const int iw = (int)((blockIdx.x * blockDim.x + threadIdx.x) >> 5);
  const int f = threadIdx.x & 31;
  const int i = __builtin_amdgcn_readfirstlane(iw);
  if (i >= n) return;
  const int g = __builtin_amdgcn_readfirstlane(batch[i]);
  const float v = act[(long)i * HID + f];
  atomicMax((unsigned int*)&gmax[(long)g * HID + f], __float_as_uint(v));
  atomicAdd(&gsum[(long)g * HID + f], v);
  if (f == 0) atomicAdd(&cnt[g], 1.0f);
}

// ---------------- final: out[g] = concat(gmax, gmean) @ W_out + b_out ----------------
__global__ void final_kernel(const float* __restrict__ gmax, const float* __restrict__ gsum,
                             const float* __restrict__ cnt, const float* __restrict__ Wout,
                             const float* __restrict__ bout, float* __restrict__ out, int g_n) {
  const int g = blockIdx.x * blockDim.x + threadIdx.x;
  if (g >= g_n) return;
  const float c = fmaxf(cnt[g], 1.0f);
  float acc = bout[0];
#pragma unroll
  for (int j = 0; j < HID; ++j) acc += gmax[(long)g * HID + j] * Wout[j];
#pragma unroll
  for (int j = 0; j < HID; ++j) acc += (gsum[(long)g * HID + j] / c) * Wout[HID + j];
  out[g] = acc;
}

// ---------------- host-side orchestration ----------------
static inline void zero_async(float* p, long n, hipStream_t s) {
  zero_kernel<<<2048, 256, 0, s>>>((float4*)p, n / 4);
}

extern "C" void kernel_launch(void* const* d_in, const int* in_sizes, int n_in,
                              void* d_out, int out_size, void* d_ws, size_t ws_size,
                              hipStream_t stream) {
  const float* x     = (const float*)d_in[0];
  const int*   eidx  = (const int*)d_in[1];  // [2, E] flat
  const int*   batch = (const int*)d_in[2];
  const float* W_in  = (const float*)d_in[3];
  const float* b_in  = (const float*)d_in[4];
  const float* W1    = (const float*)d_in[5];
  const float* b1    = (const float*)d_in[6];
  const float* W2    = (const float*)d_in[7];
  const float* b2    = (const float*)d_in[8];
  const float* W3    = (const float*)d_in[9];
  const float* b3    = (const float*)d_in[10];
  const float* W_out = (const float*)d_in[11];
  const float* b_out = (const float*)d_in[12];
  float* out = (float*)d_out;

  const int* row = eidx;
  const int* col = eidx + N_EDGES;

  // workspace layout (floats)
  float* ws   = (float*)d_ws;
  float* dinv = ws;                           // N (deg -> dinv in place)
  float* B0   = dinv + N_NODES;               // N*HID
  float* B1   = B0 + (long)N_NODES * HID;     // N*HID
  float* Hlin = B1 + (long)N_NODES * HID;     // N*HID
  float* gmax = Hlin + (long)N_NODES * HID;   // G*HID
  float* gsum = gmax + (long)N_GRAPHS * HID;  // G*HID
  float* cnt  = gsum + (long)N_GRAPHS * HID;  // G (zeroed as 512 floats)

  const int TB = 256;
  const int edge_blocks    = (N_EDGES + TB - 1) / TB;                       // 1 thread/edge
  const int edge_wv_blocks = (int)(((long)N_EDGES * 32 + TB - 1) / TB);     // 1 wave/edge
  const int node_blocks    = (N_NODES + TB - 1) / TB;
  const int nodeq_blocks   = (int)(((long)N_NODES * (HID / 4) + TB - 1) / TB);
  const int node_wv_blocks = (int)(((long)N_NODES * 32 + TB - 1) / TB);
  const int gemm_blocks    = ((N_NODES / 16) * 32 + TB - 1) / TB;           // waves*32 threads

  // degrees -> dinv
  zero_async(dinv, N_NODES, stream);
  degree_kernel<<<edge_blocks, TB, 0, stream>>>(col, dinv, N_EDGES);
  dinv_kernel<<<node_blocks, TB, 0, stream>>>(dinv, N_NODES);

  // layer 1: x[N,128] @ W_in -> Hlin; agg into B0; act in place -> B0
  gemm_wmma_kernel<F_INPUT><<<gemm_blocks, TB, 0, stream>>>(x, W_in, Hlin, N_NODES);
  zero_async(B0, (long)N_NODES * HID, stream);
  edge_scatter_kernel<<<edge_wv_blocks, TB, 0, stream>>>(row, col, dinv, Hlin, B0, N_EDGES);
  act_kernel<<<nodeq_blocks, TB, 0, stream>>>((const float4*)Hlin, dinv, (const float4*)b_in,
                                              (float4*)B0, N_NODES);

  // layer 2: B0 @ W1 -> Hlin; agg into B1
  gemm_wmma_kernel<HID><<<gemm_blocks, TB, 0, stream>>>(B0, W1, Hlin, N_NODES);
  zero_async(B1, (long)N_NODES * HID, stream);
  edge_scatter_kernel<<<edge_wv_blocks, TB, 0, stream>>>(row, col, dinv, Hlin, B1, N_EDGES);
  act_kernel<<<nodeq_blocks, TB, 0, stream>>>((const float4*)Hlin, dinv, (const float4*)b1,
                                              (float4*)B1, N_NODES);

  // layer 3: B1 @ W2 -> Hlin; agg into B0
  gemm_wmma_kernel<HID><<<gemm_blocks, TB, 0, stream>>>(B1, W2, Hlin, N_NODES);
  zero_async(B0, (long)N_NODES * HID, stream);
  edge_scatter_kernel<<<edge_wv_blocks, TB, 0, stream>>>(row, col, dinv, Hlin, B0, N_EDGES);
  act_kernel<<<nodeq_blocks, TB, 0, stream>>>((const float4*)Hlin, dinv, (const float4*)b2,
                                              (float4*)B0, N_NODES);

  // layer 4: B0 @ W3 -> Hlin; agg into B1
  gemm_wmma_kernel<HID><<<gemm_blocks, TB, 0, stream>>>(B0, W3, Hlin, N_NODES);
  zero_async(B1, (long)N_NODES * HID, stream);
  edge_scatter_kernel<<<edge_wv_blocks, TB, 0, stream>>>(row, col, dinv, Hlin, B1, N_EDGES);
  act_kernel<<<nodeq_blocks, TB, 0, stream>>>((const float4*)Hlin, dinv, (const float4*)b3,
                                              (float4*)B1, N_NODES);

  // pooling
  zero_async(gmax, (long)N_GRAPHS * HID, stream);  // relu outputs >= 0: 0 is identity for max
  zero_async(gsum, (long)N_GRAPHS * HID, stream);
  zero_async(cnt, N_GRAPHS, stream);
  pool_kernel<<<node_wv_blocks, TB, 0, stream>>>(B1, batch, gmax, gsum, cnt, N_NODES);

  // readout
  final_kernel<<<(N_GRAPHS + TB - 1) / TB, TB, 0, stream>>>(gmax, gsum, cnt, W_out, b_out,
                                                            out, N_GRAPHS);
}